// STN_48421461295728
// MI455X (gfx1250) — compile-verified
//
#include <hip/hip_runtime.h>

// STN 3D grid-sample, trilinear, border padding.
// image: [B=2, C=2, D=H=W=128] f32
// ddf:   [B=2, 3,   D,H,W]     f32  (voxel-space displacement, d/h/w order)
// out:   [B=2, C=2, D,H,W]     f32
//
// Memory-bound kernel (~117 MB total traffic, ~5us floor at 23.3 TB/s HBM).
// One thread per (b,d,h,w) voxel; coordinates/weights computed once and
// shared across both channels. Streaming ddf reads and output stores use
// non-temporal hints so the gathered image stays resident in WGP$/L2;
// ddf stream is prefetched ahead via global_prefetch_b8 (gfx1250 path).

#define DIM   128
#define NVOX  (128 * 128 * 128)   // 2097152 voxels per batch
#define BATCH 2
#define CHAN  2

__global__ __launch_bounds__(256) void stn3d_trilinear_border(
    const float* __restrict__ image,
    const float* __restrict__ ddf,
    float* __restrict__ out)
{
    const int idx = blockIdx.x * 256 + threadIdx.x;      // [0, B*NVOX)
    const int b = idx >> 21;                             // NVOX = 2^21
    const int s = idx & (NVOX - 1);                      // d*H*W + h*W + w
    const int w = s & (DIM - 1);
    const int h = (s >> 7) & (DIM - 1);
    const int d = s >> 14;

    // ---- streaming ddf reads (read-once -> non-temporal) --------------
    const float* dp = ddf + (size_t)b * 3 * NVOX + s;
    // prefetch the ddf stream ~8KB ahead (gfx1250 global_prefetch_b8)
    __builtin_prefetch(dp + 2048, 0, 0);
    const float dd = __builtin_nontemporal_load(dp);
    const float dh = __builtin_nontemporal_load(dp + NVOX);
    const float dw = __builtin_nontemporal_load(dp + 2 * NVOX);

    // ---- absolute voxel coords, border clamp (match reference order) --
    const float cd = fminf(fmaxf((float)d + dd, 0.0f), (float)(DIM - 1));
    const float ch = fminf(fmaxf((float)h + dh, 0.0f), (float)(DIM - 1));
    const float cw = fminf(fmaxf((float)w + dw, 0.0f), (float)(DIM - 1));

    const float d0f = floorf(cd), h0f = floorf(ch), w0f = floorf(cw);
    const float fd = cd - d0f,   fh = ch - h0f,   fw = cw - w0f;

    const int d0 = (int)d0f, h0 = (int)h0f, w0 = (int)w0f;
    const int d1 = min(d0 + 1, DIM - 1);
    const int h1 = min(h0 + 1, DIM - 1);
    const int w1 = min(w0 + 1, DIM - 1);

    const float gd = 1.0f - fd, gh = 1.0f - fh, gw = 1.0f - fw;
    const float w000 = gd * gh * gw, w001 = gd * gh * fw;
    const float w010 = gd * fh * gw, w011 = gd * fh * fw;
    const float w100 = fd * gh * gw, w101 = fd * gh * fw;
    const float w110 = fd * fh * gw, w111 = fd * fh * fw;

    // ---- 8 corner offsets, shared by both channels --------------------
    const int dA = d0 << 14, dB = d1 << 14;   // d * H*W
    const int hA = h0 << 7,  hB = h1 << 7;    // h * W
    const int o000 = dA + hA + w0, o001 = dA + hA + w1;
    const int o010 = dA + hB + w0, o011 = dA + hB + w1;
    const int o100 = dB + hA + w0, o101 = dB + hA + w1;
    const int o110 = dB + hB + w0, o111 = dB + hB + w1;

    // ---- gather + trilinear blend for both channels -------------------
#pragma unroll
    for (int c = 0; c < CHAN; ++c) {
        const size_t base = ((size_t)(b * CHAN + c)) << 21;   // * NVOX
        const float* __restrict__ p = image + base;

        float acc;
        acc  = p[o000] * w000;
        acc  = fmaf(p[o001], w001, acc);
        acc  = fmaf(p[o010], w010, acc);
        acc  = fmaf(p[o011], w011, acc);
        acc  = fmaf(p[o100], w100, acc);
        acc  = fmaf(p[o101], w101, acc);
        acc  = fmaf(p[o110], w110, acc);
        acc  = fmaf(p[o111], w111, acc);

        // output is write-once streaming -> non-temporal store
        __builtin_nontemporal_store(acc, out + base + s);
    }
}

extern "C" void kernel_launch(void* const* d_in, const int* in_sizes, int n_in,
                              void* d_out, int out_size, void* d_ws, size_t ws_size,
                              hipStream_t stream) {
    const float* image = (const float*)d_in[0];   // [2,2,128,128,128] f32
    const float* ddf   = (const float*)d_in[1];   // [2,3,128,128,128] f32
    float* out = (float*)d_out;                   // [2,2,128,128,128] f32

    const int total_vox = BATCH * NVOX;           // 4,194,304 threads
    const int block = 256;                        // 8 wave32 waves
    const int grid = total_vox / block;           // 16,384 blocks

    stn3d_trilinear_border<<<grid, block, 0, stream>>>(image, ddf, out);
}